// CommunicativeMessagePassing_77884936946199
// MI455X (gfx1250) — compile-verified
//
#include <hip/hip_runtime.h>

#define N_ATOMS 50000
#define N_EDGES 200000
#define DV 133
#define DE 14
#define DH 256

typedef __attribute__((ext_vector_type(2))) float v2f;
typedef __attribute__((ext_vector_type(8))) float v8f;

enum { EP_RELU = 0, EP_ADD = 1, EP_FUSE = 2 };

#define LDS_STRIDE 402  // floats; 402%64=18 -> conflict-free across 16 rows; even -> 8B aligned

__device__ __forceinline__ v8f wmma4(v2f a, v2f b, v8f c) {
  return __builtin_amdgcn_wmma_f32_16x16x4_f32(false, a, false, b, (short)0, c,
                                               false, false);
}

// Out[M, 256] = epilogue( concat(A[gather], A2) @ W^T , Res )
// Block tile: 64 rows x 256 cols. 8 waves; wave w -> cols [32w, 32w+32), 4 M-subtiles.
template <int EP, int K1, int K2>
__global__ __launch_bounds__(256) void gemm_wmma_f32(
    const float* __restrict__ A, int lda, const int* __restrict__ gatherA,
    const float* __restrict__ A2, int lda2, const float* __restrict__ W,
    const float* __restrict__ Res, float* __restrict__ Out, int M) {
  constexpr int K = K1 + K2;
  constexpr int K4 = (K + 3) & ~3;
  constexpr int KMAIN = (K / 4) * 4;

  __shared__ float As[64 * LDS_STRIDE];
  const int tid = threadIdx.x;
  const int m0 = blockIdx.x * 64;

  // Stage A tile [64 x K4] into LDS, zero padded. 16 threads per row, contiguous k.
  {
    const int rsub = tid >> 4;  // 0..15
    const int kc = tid & 15;    // 0..15
#pragma unroll
    for (int g = 0; g < 4; ++g) {
      const int row = g * 16 + rsub;
      const int grow = m0 + row;
      const bool valid = grow < M;
      long srcRow = 0;
      if (valid) srcRow = gatherA ? (long)gatherA[grow] : (long)grow;
      for (int k = kc; k < K4; k += 16) {
        float val = 0.0f;
        if (valid) {
          if (k < K1)
            val = A[srcRow * lda + k];
          else if (k < K)
            val = A2[(long)grow * lda2 + (k - K1)];
        }
        As[row * LDS_STRIDE + k] = val;
      }
    }
  }
  __syncthreads();

  const int wave = tid >> 5;   // 0..7
  const int lane = tid & 31;
  const int half = lane >> 4;  // 0 -> K pair {0,1}; 1 -> K pair {2,3}
  const int l16 = lane & 15;
  const int n0 = wave * 32;
  const int koff = half << 1;

  v8f acc[4][2] = {};
  const float* W0 = W + (long)(n0 + l16) * K;
  const float* W1 = W + (long)(n0 + 16 + l16) * K;

#pragma unroll 4
  for (int kk = 0; kk < KMAIN; kk += 4) {
    const int ko = kk + koff;
    v2f b0, b1;
    b0.x = W0[ko];
    b0.y = W0[ko + 1];
    b1.x = W1[ko];
    b1.y = W1[ko + 1];
#pragma unroll
    for (int t = 0; t < 4; ++t) {
      v2f a;
      a.x = As[(t * 16 + l16) * LDS_STRIDE + ko];
      a.y = As[(t * 16 + l16) * LDS_STRIDE + ko + 1];
      acc[t][0] = wmma4(a, b0, acc[t][0]);
      acc[t][1] = wmma4(a, b1, acc[t][1]);
    }
  }
  if constexpr (KMAIN < K4) {  // single tail step for K % 4 != 0
    const int ko = KMAIN + koff;
    v2f b0 = {}, b1 = {};
    if (ko < K) {
      b0.x = W0[ko];
      b1.x = W1[ko];
    }
    if (ko + 1 < K) {
      b0.y = W0[ko + 1];
      b1.y = W1[ko + 1];
    }
#pragma unroll
    for (int t = 0; t < 4; ++t) {
      v2f a;
      a.x = As[(t * 16 + l16) * LDS_STRIDE + ko];
      a.y = As[(t * 16 + l16) * LDS_STRIDE + ko + 1];
      acc[t][0] = wmma4(a, b0, acc[t][0]);
      acc[t][1] = wmma4(a, b1, acc[t][1]);
    }
  }

  // C/D layout: VGPR r -> row (subtile*16 + r + 8*half), col (n0 + s*16 + l16)
#pragma unroll
  for (int t = 0; t < 4; ++t) {
#pragma unroll
    for (int r = 0; r < 8; ++r) {
      const int m = m0 + t * 16 + r + (half << 3);
      if (m >= M) continue;
#pragma unroll
      for (int s = 0; s < 2; ++s) {
        const long off = (long)m * DH + (n0 + s * 16 + l16);
        const float v = acc[t][s][r];
        float o;
        if (EP == EP_RELU)
          o = fmaxf(v, 0.0f);
        else if (EP == EP_ADD)
          o = Res[off] + v;
        else
          o = Res[off] + fmaxf(v, 0.0f);
        Out[off] = o;
      }
    }
  }
}

// a_sum += scatter(Hb by w); a_max via uint-bit max (Hb >= 0 always).
// One thread = 4 consecutive features of one edge.
__global__ __launch_bounds__(256) void segment_scatter(
    const float* __restrict__ Hb, const int* __restrict__ widx,
    float* __restrict__ a_sum, float* __restrict__ a_max) {
  const long idx = (long)blockIdx.x * 256 + threadIdx.x;
  if (idx >= (long)N_EDGES * (DH / 4)) return;
  const int e = (int)(idx >> 6);
  const int c = (int)(idx & 63) * 4;
  const float4 h = *(const float4*)(Hb + (long)e * DH + c);
  const long off = (long)widx[e] * DH + c;
  atomicAdd(a_sum + off + 0, h.x);
  atomicAdd(a_sum + off + 1, h.y);
  atomicAdd(a_sum + off + 2, h.z);
  atomicAdd(a_sum + off + 3, h.w);
  if (a_max) {
    unsigned int* mx = (unsigned int*)a_max + off;
    atomicMax(mx + 0, __float_as_uint(h.x));
    atomicMax(mx + 1, __float_as_uint(h.y));
    atomicMax(mx + 2, __float_as_uint(h.z));
    atomicMax(mx + 3, __float_as_uint(h.w));
  }
}

__global__ __launch_bounds__(256) void gate_mul(const float* __restrict__ a_sum,
                                                float* __restrict__ gate) {
  const long idx = ((long)blockIdx.x * 256 + threadIdx.x) * 4;
  if (idx >= (long)N_ATOMS * DH) return;
  const float4 s = *(const float4*)(a_sum + idx);
  float4 g = *(const float4*)(gate + idx);
  g.x *= s.x;
  g.y *= s.y;
  g.z *= s.z;
  g.w *= s.w;
  *(float4*)(gate + idx) = g;
}

// Mb[e, :] = a_sum[v[e], :] - Hb[rev[e], :]
__global__ __launch_bounds__(256) void compute_Mb(
    const float* __restrict__ a_sum, const float* __restrict__ Hb,
    const int* __restrict__ vidx, const int* __restrict__ rev,
    float* __restrict__ Mb) {
  const long idx = (long)blockIdx.x * 256 + threadIdx.x;
  if (idx >= (long)N_EDGES * (DH / 4)) return;
  const int e = (int)(idx >> 6);
  const int c = (int)(idx & 63) * 4;
  const float4 s = *(const float4*)(a_sum + (long)vidx[e] * DH + c);
  const float4 h = *(const float4*)(Hb + (long)rev[e] * DH + c);
  float4 m;
  m.x = s.x - h.x;
  m.y = s.y - h.y;
  m.z = s.z - h.z;
  m.w = s.w - h.w;
  *(float4*)(Mb + (long)e * DH + c) = m;
}

extern "C" void kernel_launch(void* const* d_in, const int* in_sizes, int n_in,
                              void* d_out, int out_size, void* d_ws,
                              size_t ws_size, hipStream_t stream) {
  const float* V       = (const float*)d_in[0];
  const float* E_feat  = (const float*)d_in[1];
  const int*   eidx    = (const int*)d_in[2];  // [2, E]
  const int*   rev     = (const int*)d_in[3];
  const float* Wi_atom = (const float*)d_in[4];
  const float* Wi_bond = (const float*)d_in[5];
  const float* Wh_atom = (const float*)d_in[6];
  const float* Wh_bond = (const float*)d_in[7];
  const float* Wf_atom = (const float*)d_in[8];
  const float* Wf_bond = (const float*)d_in[9];
  const float* Wo_atom = (const float*)d_in[10];

  const int* v_idx = eidx;
  const int* w_idx = eidx + N_EDGES;

  const long NA = (long)N_ATOMS * DH;
  const long NE = (long)N_EDGES * DH;
  float* Ha    = (float*)d_ws;
  float* Hb    = Ha + NA;
  float* a_sum = Hb + NE;
  float* gate  = a_sum + NA;  // doubles as a_max
  float* tmpA  = gate + NA;
  float* tmpB  = tmpA + NA;   // M_b
  float* tmpB2 = tmpB + NE;   // S_b
  float* Hout  = (float*)d_out;

  const dim3 blk(256);
  const int gA = (N_ATOMS + 63) / 64;
  const int gE = (N_EDGES + 63) / 64;
  const int eEW = (int)((NE / 4 + 255) / 256);
  const int eAW = (int)((NA / 4 + 255) / 256);

  // H_a = relu(V @ Wi_atom^T)
  gemm_wmma_f32<EP_RELU, DV, 0><<<gA, blk, 0, stream>>>(
      V, DV, nullptr, nullptr, 0, Wi_atom, nullptr, Ha, N_ATOMS);
  // H_b = relu(concat(V[v], E_feat) @ Wi_bond^T)
  gemm_wmma_f32<EP_RELU, DV, DE><<<gE, blk, 0, stream>>>(
      V, DV, v_idx, E_feat, DE, Wi_bond, nullptr, Hb, N_EDGES);

  for (int it = 0; it < 2; ++it) {
    hipMemsetAsync(a_sum, 0, NA * sizeof(float), stream);
    hipMemsetAsync(gate, 0, NA * sizeof(float), stream);
    segment_scatter<<<eEW, blk, 0, stream>>>(Hb, w_idx, a_sum, gate);
    gate_mul<<<eAW, blk, 0, stream>>>(a_sum, gate);

    // S_a = Ha + gate @ Wh_atom^T ; Ha = Ha + relu(S_a @ Wf_atom^T)
    gemm_wmma_f32<EP_ADD, DH, 0><<<gA, blk, 0, stream>>>(
        gate, DH, nullptr, nullptr, 0, Wh_atom, Ha, tmpA, N_ATOMS);
    gemm_wmma_f32<EP_FUSE, DH, 0><<<gA, blk, 0, stream>>>(
        tmpA, DH, nullptr, nullptr, 0, Wf_atom, Ha, Ha, N_ATOMS);

    // M_b = a_sum[v] - Hb[rev] ; S_b = Hb + M_b @ Wh_bond^T ; Hb += relu(S_b @ Wf_bond^T)
    compute_Mb<<<eEW, blk, 0, stream>>>(a_sum, Hb, v_idx, rev, tmpB);
    gemm_wmma_f32<EP_ADD, DH, 0><<<gE, blk, 0, stream>>>(
        tmpB, DH, nullptr, nullptr, 0, Wh_bond, Hb, tmpB2, N_EDGES);
    gemm_wmma_f32<EP_FUSE, DH, 0><<<gE, blk, 0, stream>>>(
        tmpB2, DH, nullptr, nullptr, 0, Wf_bond, Hb, Hb, N_EDGES);
  }

  // Final: a_sum only, then H_atom = relu(concat(V, a_sum) @ Wo_atom^T)
  hipMemsetAsync(a_sum, 0, NA * sizeof(float), stream);
  segment_scatter<<<eEW, blk, 0, stream>>>(Hb, w_idx, a_sum, nullptr);
  gemm_wmma_f32<EP_RELU, DV, DH><<<gA, blk, 0, stream>>>(
      V, DV, nullptr, a_sum, DH, Wo_atom, nullptr, Hout, N_ATOMS);
}